// GeometricReconstructionLoss_80805514707373
// MI455X (gfx1250) — compile-verified
//
#include <hip/hip_runtime.h>
#include <hip/hip_bf16.h>

typedef __attribute__((ext_vector_type(2))) float v2f;
typedef __attribute__((ext_vector_type(8))) float v8f;

// Constants for this problem instance: B=8, V=4, N=2048, D=3.
#define NPTS    2048
#define NTILE   (NPTS / 16)         // 128 reference tiles per sweep
#define NPAIR   32                  // B*V
#define NPART   (NPAIR * 2 * NTILE) // 8192 partial sums
#define NCLOUDP (NPAIR * NPTS)      // 65536 points per input tensor

// ---------------------------------------------------------------------------
// Pre-pass: build augmented points (x, y, z, |p|^2) for both input tensors.
// aug is float4[2][NPAIR][NPTS]; [0] = Xv clouds, [1] = Tv clouds.
// ---------------------------------------------------------------------------
__global__ __launch_bounds__(256)
void chamfer_prep_kernel(const float* __restrict__ Xv,
                         const float* __restrict__ Tv,
                         float4* __restrict__ aug) {
    const int i = blockIdx.x * 256 + threadIdx.x;   // 0 .. 2*NCLOUDP-1
    const float* src = (i < NCLOUDP) ? Xv : Tv;
    const int j = i & (NCLOUDP - 1);
    const float x = src[3 * j + 0];
    const float y = src[3 * j + 1];
    const float z = src[3 * j + 2];
    float4 o;
    o.x = x; o.y = y; o.z = z;
    o.w = x * x + y * y + z * z;
    aug[i] = o;
}

// ---------------------------------------------------------------------------
// Fast path: one wave per block, 16 query points vs all 2048 reference points.
// The WMMA's padded K=3 slot carries the row norm:
//   A row m = (r_x, r_y, r_z, |r_m|^2),  B col n = (-2q_x, -2q_y, -2q_z, 1)
//   => C[m][n] = |r_m|^2 - 2 * (r_m . q_n)
// Per-lane A operand is a single aligned b64 load from the aug table:
// lane L reads aug[point].xy (half 0) or aug[point].zw (half 1).
// ---------------------------------------------------------------------------
__global__ __launch_bounds__(32)
void chamfer_min_aug_kernel(const float4* __restrict__ aug,
                            float* __restrict__ partial) {
    const int bid   = blockIdx.x;
    const int chunk = bid & (NTILE - 1);        // query tile index
    const int dir   = (bid >> 7) & 1;           // 0: x->y (cham_x), 1: y->x
    const int p     = bid >> 8;                 // (b,v) pair, p = b*4 + v
    const int view  = p & 3;

    const float4* QA = aug + (dir ? NCLOUDP : 0) + (size_t)p * NPTS;
    const float4* RA = aug + (dir ? 0 : NCLOUDP) + (size_t)p * NPTS;

    const int lane = threadIdx.x;
    const int l15  = lane & 15;
    const int half = lane >> 4;

    // ---- Query point -> B operand (4x16, K split across half-waves) --------
    const float4 qv = QA[chunk * 16 + l15];
    const float qq = qv.w;

    v2f b;
    b.x = half ? (-2.0f * qv.z) : (-2.0f * qv.x);   // K = 2*half
    b.y = half ? 1.0f           : (-2.0f * qv.y);   // K = 2*half + 1

    // Per-lane pointer into the aug table: v2f element (2 per point).
    const v2f* ra = (const v2f*)RA + 2 * l15 + half;

    float vmin0 = 3.4e38f;
    float vmin1 = 3.4e38f;

    for (int t = 0; t < NTILE; ++t) {
        // A operand for row M = l15: (x,y) on half 0, (z,|r|^2) on half 1.
        const v2f a = ra[t * 32];       // stride 16 points * 2 v2f

        v8f c = {};
        // c[m][n] = |r_m|^2 - 2 * r_m . q_n
        c = __builtin_amdgcn_wmma_f32_16x16x4_f32(
                /*neg_a=*/false, a, /*neg_b=*/false, b,
                /*c_mod=*/(short)0, c, /*reuse_a=*/false, /*reuse_b=*/false);

        // Lane holds column n = l15, rows M = 8*half + j. Two min chains.
        vmin0 = fminf(fminf(vmin0, c[0]), c[1]);
        vmin1 = fminf(fminf(vmin1, c[2]), c[3]);
        vmin0 = fminf(fminf(vmin0, c[4]), c[5]);
        vmin1 = fminf(fminf(vmin1, c[6]), c[7]);
    }

    float vmin = fminf(vmin0, vmin1);
    // Merge the two half-wave row ranges (rows 0-7 vs 8-15 of every tile).
    vmin = fminf(vmin, __shfl_xor(vmin, 16, 32));
    // Full nearest-neighbor squared distance for query point n = l15.
    float s = vmin + qq;

    // Sum the 16 query mins (butterfly over lanes 0..15; lanes 16..31 dup).
#pragma unroll
    for (int off = 1; off < 16; off <<= 1)
        s += __shfl_xor(s, off, 32);

    if (lane == 0) {
        // final = sum_{b,v} w_v/B * (cham_x + cham_y),  w_v = 2 if v==0 else 1
        const float w = ((view == 0) ? 2.0f : 1.0f) * (1.0f / 8.0f);
        partial[bid] = w * s;
    }
}

// ---------------------------------------------------------------------------
// Fallback path (small ws): raw loads, rr computed in-loop (round-2 version).
// ---------------------------------------------------------------------------
__global__ __launch_bounds__(32)
void chamfer_min_raw_kernel(const float* __restrict__ Xv,
                            const float* __restrict__ Tv,
                            float* __restrict__ partial) {
    const int bid   = blockIdx.x;
    const int chunk = bid & (NTILE - 1);
    const int dir   = (bid >> 7) & 1;
    const int p     = bid >> 8;
    const int view  = p & 3;

    const size_t base = (size_t)p * NPTS * 3;
    const float* Q = dir ? (Tv + base) : (Xv + base);
    const float* R = dir ? (Xv + base) : (Tv + base);

    const int lane = threadIdx.x;
    const int l15  = lane & 15;
    const int half = lane >> 4;

    const int q = chunk * 16 + l15;
    const float qx = Q[3 * q + 0];
    const float qy = Q[3 * q + 1];
    const float qz = Q[3 * q + 2];
    const float qq = qx * qx + qy * qy + qz * qz;

    v2f b;
    b.x = half ? (-2.0f * qz) : (-2.0f * qx);
    b.y = half ? 1.0f         : (-2.0f * qy);

    float vmin = 3.4e38f;

    for (int t = 0; t < NTILE; ++t) {
        const float* rp = R + 3 * (t * 16 + l15);
        const float rx = rp[0];
        const float ry = rp[1];
        const float rz = rp[2];
        const float rr = rx * rx + ry * ry + rz * rz;

        v2f a;
        a.x = half ? rz : rx;
        a.y = half ? rr : ry;

        v8f c = {};
        c = __builtin_amdgcn_wmma_f32_16x16x4_f32(
                false, a, false, b, (short)0, c, false, false);

#pragma unroll
        for (int j = 0; j < 8; ++j)
            vmin = fminf(vmin, c[j]);
    }

    vmin = fminf(vmin, __shfl_xor(vmin, 16, 32));
    float s = vmin + qq;
#pragma unroll
    for (int off = 1; off < 16; off <<= 1)
        s += __shfl_xor(s, off, 32);

    if (lane == 0) {
        const float w = ((view == 0) ? 2.0f : 1.0f) * (1.0f / 8.0f);
        partial[bid] = w * s;
    }
}

// Deterministic fixed-order reduction of the 8192 partials.
__global__ __launch_bounds__(256)
void chamfer_reduce_kernel(const float* __restrict__ partial,
                           float* __restrict__ out) {
    __shared__ float sm[256];
    const int tid = threadIdx.x;
    float s = 0.f;
    for (int k = tid; k < NPART; k += 256)
        s += partial[k];
    sm[tid] = s;
    __syncthreads();
    for (int off = 128; off > 0; off >>= 1) {
        if (tid < off) sm[tid] += sm[tid + off];
        __syncthreads();
    }
    if (tid == 0) out[0] = sm[0];
}

extern "C" void kernel_launch(void* const* d_in, const int* in_sizes, int n_in,
                              void* d_out, int out_size, void* d_ws, size_t ws_size,
                              hipStream_t stream) {
    const float* Xv = (const float*)d_in[0];
    const float* Tv = (const float*)d_in[1];
    float* out      = (float*)d_out;

    const size_t augBytes = (size_t)2 * NCLOUDP * sizeof(float4);  // 2 MB

    if (ws_size >= augBytes + (size_t)NPART * sizeof(float)) {
        // Fast path: augmented-point table in workspace.
        float4* aug    = (float4*)d_ws;
        float* partial = (float*)((char*)d_ws + augBytes);

        chamfer_prep_kernel<<<dim3(2 * NCLOUDP / 256), dim3(256), 0, stream>>>(
            Xv, Tv, aug);
        chamfer_min_aug_kernel<<<dim3(NPART), dim3(32), 0, stream>>>(
            aug, partial);
        chamfer_reduce_kernel<<<dim3(1), dim3(256), 0, stream>>>(partial, out);
    } else {
        // Fallback: raw loads, only 32 KB of workspace needed.
        float* partial = (float*)d_ws;
        chamfer_min_raw_kernel<<<dim3(NPART), dim3(32), 0, stream>>>(
            Xv, Tv, partial);
        chamfer_reduce_kernel<<<dim3(1), dim3(256), 0, stream>>>(partial, out);
    }
}